// FullAttention_v53_47330539601998
// MI455X (gfx1250) — compile-verified
//
#include <hip/hip_runtime.h>
#include <math.h>

typedef __attribute__((ext_vector_type(2))) float v2f;
typedef __attribute__((ext_vector_type(8))) float v8f;

#define N_STATIONS 50
#define B_TOT 3200
#define L_DIM 7
#define H_DIM 8
#define E_DIM 64
#define S_DIM 56
#define D_DIM 64

#define WAVES_PER_BLOCK 8
#define LDS_STRIDE 68   // 64 + 4 pad -> conflict-free D-layout store & b64 A-frag read

// One wave32 per (b, h) pair. fp32 end-to-end via V_WMMA_F32_16X16X4_F32.
// All padding handled by index CLAMPING (in-bounds reads, garbage confined to
// rows/cols that are softmax-zeroed or never stored) -> no EXEC predication in
// the hot loops; every load is base + immediate offset after full unroll.
__global__ __launch_bounds__(256, 2)
void attn_f32_wmma_kernel(const float* __restrict__ Q,
                          const float* __restrict__ K,
                          const float* __restrict__ V,
                          const float* __restrict__ Mask,
                          float* __restrict__ O)
{
    __shared__ __align__(16) float ldsP[WAVES_PER_BLOCK * 16 * LDS_STRIDE];

    const int tid  = threadIdx.x;
    const int wave = tid >> 5;
    const int lane = tid & 31;
    const int m    = lane & 15;   // row (A/D) or column (B) index inside a 16-tile
    const int half = lane >> 4;   // 0: K sub-cols {0,1}, 1: {2,3}

    const int pair = blockIdx.x * WAVES_PER_BLOCK + wave;  // exact: 3200*8 pairs
    const int b  = pair >> 3;     // / H_DIM
    const int h  = pair & 7;      // % H_DIM
    const int st = b % N_STATIONS;

    float* lds = ldsP + wave * (16 * LDS_STRIDE);

    // ---- clamped per-lane base pointers (hoisted; no predication needed) ----
    const int qrow = (m < L_DIM) ? m : (L_DIM - 1);          // pad rows re-read row 6
    const float* Qrow = Q + ((size_t)(b * L_DIM + qrow) * H_DIM + h) * E_DIM + 2 * half;

    const float* Krow[4];
#pragma unroll
    for (int t = 0; t < 4; ++t) {
        int s = 16 * t + m;
        if (s > S_DIM - 1) s = S_DIM - 1;                    // pad cols re-read s=55
        Krow[t] = K + ((size_t)(b * S_DIM + s) * H_DIM + h) * E_DIM + 2 * half;
    }

    const float* Vcol[4];
#pragma unroll
    for (int t = 0; t < 4; ++t)
        Vcol[t] = V + ((size_t)b * S_DIM + 2 * half) * (H_DIM * D_DIM)
                    + h * D_DIM + 16 * t + m;

    const v8f vzero = {0.f,0.f,0.f,0.f,0.f,0.f,0.f,0.f};

    // ---------------- GEMM1: scores[16 x 64span] = Q @ K^T ----------------
    v8f acc[4] = {vzero, vzero, vzero, vzero};

#pragma unroll
    for (int kk = 0; kk < 16; ++kk) {
        v2f a = *(const v2f*)(Qrow + 4 * kk);                // imm offset 16*kk B
#pragma unroll
        for (int t = 0; t < 4; ++t) {
            v2f bf = *(const v2f*)(Krow[t] + 4 * kk);
            acc[t] = __builtin_amdgcn_wmma_f32_16x16x4_f32(
                false, a, false, bf, (short)0, acc[t], false, false);
        }
    }

    // -------- mask * score, scale, softmax over s; write probs to LDS --------
    const float scale = 0.125f;  // 1/sqrt(E)
    const float* Mst = Mask + (size_t)st * (L_DIM * S_DIM);

#pragma unroll
    for (int r = 0; r < L_DIM; ++r) {        // r=7 rows are all-invalid: zero-filled below
        const int l  = r + 8 * half;         // D-layout: VGPR r = row r / r+8
        const int lc = (l < L_DIM) ? l : (L_DIM - 1);
        const float* Mrow = Mst + lc * S_DIM;
        float vv[4];
#pragma unroll
        for (int t = 0; t < 4; ++t) {
            const int s  = 16 * t + m;
            const int sc = (s < S_DIM) ? s : (S_DIM - 1);
            const float mval = Mrow[sc];                     // unconditional, in-bounds
            const bool valid = (l < L_DIM) && (s < S_DIM);
            vv[t] = valid ? acc[t][r] * mval * scale : -INFINITY;
        }
        // row reduction across the 16 lanes of this half (xor <=8 stays in-half)
        float mx = fmaxf(fmaxf(vv[0], vv[1]), fmaxf(vv[2], vv[3]));
        for (int off = 8; off >= 1; off >>= 1) mx = fmaxf(mx, __shfl_xor(mx, off, 32));

        float e[4];
        float ssum = 0.f;
#pragma unroll
        for (int t = 0; t < 4; ++t) {
            e[t] = (vv[t] == -INFINITY) ? 0.f : __expf(vv[t] - mx);
            ssum += e[t];
        }
        for (int off = 8; off >= 1; off >>= 1) ssum += __shfl_xor(ssum, off, 32);
        const float inv = (ssum > 0.f) ? (1.0f / ssum) : 0.f;

#pragma unroll
        for (int t = 0; t < 4; ++t)
            lds[(size_t)l * LDS_STRIDE + 16 * t + m] = e[t] * inv;
    }
    // zero the two untouched pad rows (7 and 15) so GEMM2 pad rows are clean zeros
#pragma unroll
    for (int t = 0; t < 4; ++t)
        lds[(size_t)(L_DIM + 8 * half) * LDS_STRIDE + 16 * t + m] = 0.f;

    // ---------------- GEMM2: out[16 x 64] = P(16 x 56) @ V(56 x 64) ----------------
    v8f acc2[4] = {vzero, vzero, vzero, vzero};
    const float* ldsA = lds + (size_t)m * LDS_STRIDE + 2 * half;

#pragma unroll
    for (int kk = 0; kk < 14; ++kk) {        // S=56 = 14 * 4, exact (no K padding)
        v2f a = *(const v2f*)(ldsA + 4 * kk);                // ds_load_b64, imm 16*kk
#pragma unroll
        for (int t = 0; t < 4; ++t) {
            v2f bf;
            bf.x = Vcol[t][(size_t)(4 * kk)     * (H_DIM * D_DIM)];  // imm 8192*kk B
            bf.y = Vcol[t][(size_t)(4 * kk + 1) * (H_DIM * D_DIM)];  // imm +2048 B
            acc2[t] = __builtin_amdgcn_wmma_f32_16x16x4_f32(
                false, a, false, bf, (short)0, acc2[t], false, false);
        }
    }

    // ---------------- store out[b, l, h, d], valid rows l = 0..6 ----------------
    if (half == 0) {
#pragma unroll
        for (int r = 0; r < L_DIM; ++r) {
            const size_t base = ((size_t)(b * L_DIM + r) * H_DIM + h) * D_DIM;
#pragma unroll
            for (int t = 0; t < 4; ++t)
                O[base + 16 * t + m] = acc2[t][r];
        }
    }
}

extern "C" void kernel_launch(void* const* d_in, const int* in_sizes, int n_in,
                              void* d_out, int out_size, void* d_ws, size_t ws_size,
                              hipStream_t stream) {
    const float* queries = (const float*)d_in[0];  // [B, L, H, E]
    const float* keys    = (const float*)d_in[1];  // [B, S, H, E]
    const float* values  = (const float*)d_in[2];  // [B, S, H, D]
    const float* mask    = (const float*)d_in[3];  // [N, L, S]
    float* out = (float*)d_out;                    // [B, L, H, D]

    const int pairs  = B_TOT * H_DIM;              // 25600 waves
    const int blocks = pairs / WAVES_PER_BLOCK;    // 3200 blocks of 256 threads
    attn_f32_wmma_kernel<<<blocks, 256, 0, stream>>>(queries, keys, values, mask, out);
}